// ComplexMultiheadAttention_81647328297518
// MI455X (gfx1250) — compile-verified
//
#include <hip/hip_runtime.h>

typedef __attribute__((ext_vector_type(2))) float v2f;
typedef __attribute__((ext_vector_type(8))) float v8f;

#define B_  4
#define S_  1024
#define D_  512
#define H_  8
#define DH_ 64
#define EPS_ 1e-5f

// ---------------------------------------------------------------------------
// WMMA helper: D = A(16x4) * B(4x16) + C, all f32. wave32.
// 8-arg form: (neg_a, A, neg_b, B, c_mod, C, reuse_a, reuse_b)
// ---------------------------------------------------------------------------
__device__ __forceinline__ v8f wmma_f32(v2f a, v2f b, v8f c) {
    return __builtin_amdgcn_wmma_f32_16x16x4_f32(false, a, false, b, (short)0, c,
                                                 false, false);
}

__device__ __forceinline__ v2f lds2(const float* p) {
    return *(const v2f*)p;   // even-index offsets only -> 8B aligned ds_load_b64
}

// ---------------------------------------------------------------------------
// CDNA5 async global->LDS copy (ASYNCcnt path), b64 granularity.
// LDS flat address: low 32 bits of the generic pointer are the LDS offset.
// ---------------------------------------------------------------------------
__device__ __forceinline__ void async_b64(unsigned lds_off, const float* g) {
    asm volatile("global_load_async_to_lds_b64 %0, %1, off"
                 :: "v"(lds_off), "v"(g)
                 : "memory");
}
__device__ __forceinline__ void wait_async0() {
    asm volatile("s_wait_asynccnt 0x0" ::: "memory");
}

// Stage one 32x64 f32 plane into LDS with b64-chunk XOR swizzle:
// element (row, 2c..2c+1) lives at lds[row*64 + 2*(c ^ row)].
// 128 threads x 8 chunks = 1024 b64 chunks.
__device__ __forceinline__ void stage_plane_async(const float* __restrict__ gsrc,
                                                  float* lds_base, int t)
{
#pragma unroll
    for (int j = 0; j < 8; ++j) {
        const int c   = t * 8 + j;
        const int row = c >> 5;
        const int ch  = c & 31;
        const unsigned off =
            (unsigned)(size_t)(lds_base + row * 64 + 2 * (ch ^ row));
        async_b64(off, gsrc + (size_t)row * D_ + ch * 2);
    }
}

// ---------------------------------------------------------------------------
// Kernel 1: complex projection  Y[m,e] = sum_d X[m,d]*W[e,d] + b[e]
// X: [M, D, 2] interleaved.  W: [D(out e), D(in d), 2].  b: [D,2]
// Output planar: Yr[M,D], Yi[M,D]
// Block tile 128(M) x 64(N), 8 waves, each wave 2x2 subtiles of 16x16.
// (Staging goes through VGPRs because the complex de-interleave needs it.)
// ---------------------------------------------------------------------------
#define PJ_LDA 18   // 16 K + pad 2 (even -> aligned b64, conflict-free)

__global__ __launch_bounds__(256)
void proj_kernel(const float* __restrict__ X, const float* __restrict__ W,
                 const float* __restrict__ bias,
                 float* __restrict__ Yr, float* __restrict__ Yi)
{
    __shared__ float Ar[128 * PJ_LDA], Ai[128 * PJ_LDA];
    __shared__ float Br[64 * PJ_LDA],  Bi[64 * PJ_LDA];

    const int t    = threadIdx.x;
    const int w    = t >> 5;
    const int lane = t & 31;
    const int half = lane >> 4;
    const int lrow = lane & 15;
    const int wm   = w & 3;        // 4 M-subblocks of 32
    const int wn   = w >> 2;       // 2 N-subblocks of 32
    const int m0   = blockIdx.x * 128;
    const int n0   = blockIdx.y * 64;

    v8f accR[2][2], accI[2][2];
#pragma unroll
    for (int mi = 0; mi < 2; ++mi)
#pragma unroll
        for (int ni = 0; ni < 2; ++ni) {
            accR[mi][ni] = (v8f){0,0,0,0,0,0,0,0};
            accI[mi][ni] = (v8f){0,0,0,0,0,0,0,0};
        }

    for (int kp = 0; kp < D_ / 16; ++kp) {
        const int k0 = kp * 16;
        // ---- stage A (X): 128 rows x 16 complex; 8 complex per thread
        {
            const int row = t >> 1, seg = t & 1;
            const float4* src =
                (const float4*)(X + ((size_t)(m0 + row) * D_ + k0 + seg * 8) * 2);
            float* ar = Ar + row * PJ_LDA + seg * 8;
            float* ai = Ai + row * PJ_LDA + seg * 8;
#pragma unroll
            for (int j = 0; j < 4; ++j) {
                float4 vv = src[j];
                ar[j * 2 + 0] = vv.x;  ai[j * 2 + 0] = vv.y;
                ar[j * 2 + 1] = vv.z;  ai[j * 2 + 1] = vv.w;
            }
        }
        // ---- stage B (W, transposed view): 64 e-rows x 16 complex
        {
            const int n = t >> 2, seg = t & 3;
            const float4* src =
                (const float4*)(W + ((size_t)(n0 + n) * D_ + k0 + seg * 4) * 2);
            float* br = Br + n * PJ_LDA + seg * 4;
            float* bi = Bi + n * PJ_LDA + seg * 4;
#pragma unroll
            for (int j = 0; j < 2; ++j) {
                float4 vv = src[j];
                br[j * 2 + 0] = vv.x;  bi[j * 2 + 0] = vv.y;
                br[j * 2 + 1] = vv.z;  bi[j * 2 + 1] = vv.w;
            }
        }
        __syncthreads();

#pragma unroll
        for (int ks = 0; ks < 4; ++ks) {
            const int kc = ks * 4 + half * 2;
            v2f ar[2], ai[2], an[2], br[2], bi[2];
#pragma unroll
            for (int mi = 0; mi < 2; ++mi) {
                const int row = (wm * 32 + mi * 16 + lrow) * PJ_LDA + kc;
                ar[mi] = lds2(Ar + row);
                ai[mi] = lds2(Ai + row);
                an[mi] = -ai[mi];
            }
#pragma unroll
            for (int ni = 0; ni < 2; ++ni) {
                const int row = (wn * 32 + ni * 16 + lrow) * PJ_LDA + kc;
                br[ni] = lds2(Br + row);
                bi[ni] = lds2(Bi + row);
            }
#pragma unroll
            for (int mi = 0; mi < 2; ++mi)
#pragma unroll
                for (int ni = 0; ni < 2; ++ni) {
                    accR[mi][ni] = wmma_f32(ar[mi], br[ni], accR[mi][ni]);
                    accR[mi][ni] = wmma_f32(an[mi], bi[ni], accR[mi][ni]);
                    accI[mi][ni] = wmma_f32(ar[mi], bi[ni], accI[mi][ni]);
                    accI[mi][ni] = wmma_f32(ai[mi], br[ni], accI[mi][ni]);
                }
        }
        __syncthreads();
    }

    // ---- bias + planar writeback
#pragma unroll
    for (int mi = 0; mi < 2; ++mi)
#pragma unroll
        for (int ni = 0; ni < 2; ++ni) {
            const int e = n0 + wn * 32 + ni * 16 + lrow;
            const float bre = bias[e * 2 + 0];
            const float bim = bias[e * 2 + 1];
#pragma unroll
            for (int r = 0; r < 8; ++r) {
                const int m = m0 + wm * 32 + mi * 16 + r + half * 8;
                Yr[(size_t)m * D_ + e] = accR[mi][ni][r] + bre;
                Yi[(size_t)m * D_ + e] = accI[mi][ni][r] + bim;
            }
        }
}

// ---------------------------------------------------------------------------
// Kernel 2: flash-style complex attention per (b, h, 32-query block).
// Q/K/V tiles staged HBM->LDS via GLOBAL_LOAD_ASYNC_TO_LDS_B64 (ASYNCcnt),
// XOR-swizzled packed layout. Two passes over key tiles: pass1 row-max,
// pass2 exp/sum + P.V via WMMA (4 accumulator streams: real/imag softmax
// normalizers are independent).
// ---------------------------------------------------------------------------
#define QB  32
#define KB  32
#define PLD 34   // 32 keys + pad 2 (P buffer is written by VALU, so padded)

__global__ __launch_bounds__(128)
void attn_kernel(const float* __restrict__ Qr, const float* __restrict__ Qi,
                 const float* __restrict__ Kr, const float* __restrict__ Ki,
                 const float* __restrict__ Vr, const float* __restrict__ Vi,
                 float* __restrict__ Og_r, float* __restrict__ Og_i)
{
    __shared__ float Qs_r[QB * 64], Qs_i[QB * 64];   // swizzled [row][64]
    __shared__ float KVr[KB * 64],  KVi[KB * 64];    // K tile, then V tile
    __shared__ float Pr[QB * PLD],  Pi[QB * PLD];
    __shared__ float rmax_r[QB], rmax_i[QB], rsum_r[QB], rsum_i[QB];

    const int t    = threadIdx.x;
    const int w    = t >> 5;
    const int lane = t & 31;
    const int half = lane >> 4;
    const int lrow = lane & 15;
    const int smt  = w & 1;    // score/P M tile (16 q rows)
    const int snt  = w >> 1;   // score N tile (16 keys) / PV dh block (x32)

    const int NQB = S_ / QB;
    const int bh  = blockIdx.x / NQB;
    const int qb  = blockIdx.x % NQB;
    const int b   = bh / H_;
    const int h   = bh % H_;
    const size_t base = (size_t)b * S_ * D_ + (size_t)h * DH_;

    // ---- async stage Q tile once
    stage_plane_async(Qr + base + (size_t)(qb * QB) * D_, Qs_r, t);
    stage_plane_async(Qi + base + (size_t)(qb * QB) * D_, Qs_i, t);
    wait_async0();
    __syncthreads();

    // ===================== pass 1: row maxima =====================
    float runmax = -3.0e38f;
    for (int kt = 0; kt < S_ / KB; ++kt) {
        stage_plane_async(Kr + base + (size_t)(kt * KB) * D_, KVr, t);
        stage_plane_async(Ki + base + (size_t)(kt * KB) * D_, KVi, t);
        wait_async0();
        __syncthreads();

        v8f Sr = (v8f){0,0,0,0,0,0,0,0}, Si = (v8f){0,0,0,0,0,0,0,0};
        const int qrow = smt * 16 + lrow;
        const int krow = snt * 16 + lrow;
#pragma unroll
        for (int ks = 0; ks < DH_ / 4; ++ks) {
            const int kl = ks * 2 + half;            // b64 chunk index
            v2f aqr = lds2(Qs_r + qrow * 64 + 2 * (kl ^ qrow));
            v2f aqi = lds2(Qs_i + qrow * 64 + 2 * (kl ^ qrow));
            v2f bkr = lds2(KVr  + krow * 64 + 2 * (kl ^ krow));
            v2f bki = lds2(KVi  + krow * 64 + 2 * (kl ^ krow));
            v2f aqn = -aqi;
            Sr = wmma_f32(aqr, bkr, Sr);
            Sr = wmma_f32(aqn, bki, Sr);
            Si = wmma_f32(aqr, bki, Si);
            Si = wmma_f32(aqi, bkr, Si);
        }
#pragma unroll
        for (int r = 0; r < 8; ++r) {
            const int m = smt * 16 + r + half * 8;
            const int c = snt * 16 + lrow;
            Pr[m * PLD + c] = Sr[r] * 0.125f;   // 1/sqrt(dh)
            Pi[m * PLD + c] = Si[r] * 0.125f;
        }
        __syncthreads();
        if (t < 2 * QB) {
            const float* src = (t < QB) ? (Pr + t * PLD) : (Pi + (t - QB) * PLD);
            float mx = runmax;
            for (int c = 0; c < KB; ++c) mx = fmaxf(mx, src[c]);
            runmax = mx;
        }
        __syncthreads();
    }
    if (t < QB)            rmax_r[t]      = runmax;
    else if (t < 2 * QB)   rmax_i[t - QB] = runmax;
    __syncthreads();

    // ===================== pass 2: exp / sums / P.V =====================
    float runsum = 0.0f;
    v8f Arr[2], Aii[2], Ari[2], Air[2];
#pragma unroll
    for (int ni = 0; ni < 2; ++ni) {
        Arr[ni] = (v8f){0,0,0,0,0,0,0,0}; Aii[ni] = (v8f){0,0,0,0,0,0,0,0};
        Ari[ni] = (v8f){0,0,0,0,0,0,0,0}; Air[ni] = (v8f){0,0,0,0,0,0,0,0};
    }

    for (int kt = 0; kt < S_ / KB; ++kt) {
        stage_plane_async(Kr + base + (size_t)(kt * KB) * D_, KVr, t);
        stage_plane_async(Ki + base + (size_t)(kt * KB) * D_, KVi, t);
        wait_async0();
        __syncthreads();

        v8f Sr = (v8f){0,0,0,0,0,0,0,0}, Si = (v8f){0,0,0,0,0,0,0,0};
        const int qrow = smt * 16 + lrow;
        const int krow = snt * 16 + lrow;
#pragma unroll
        for (int ks = 0; ks < DH_ / 4; ++ks) {
            const int kl = ks * 2 + half;
            v2f aqr = lds2(Qs_r + qrow * 64 + 2 * (kl ^ qrow));
            v2f aqi = lds2(Qs_i + qrow * 64 + 2 * (kl ^ qrow));
            v2f bkr = lds2(KVr  + krow * 64 + 2 * (kl ^ krow));
            v2f bki = lds2(KVi  + krow * 64 + 2 * (kl ^ krow));
            v2f aqn = -aqi;
            Sr = wmma_f32(aqr, bkr, Sr);
            Sr = wmma_f32(aqn, bki, Sr);
            Si = wmma_f32(aqr, bki, Si);
            Si = wmma_f32(aqi, bkr, Si);
        }
#pragma unroll
        for (int r = 0; r < 8; ++r) {
            const int m = smt * 16 + r + half * 8;
            const int c = snt * 16 + lrow;
            Pr[m * PLD + c] = Sr[r] * 0.125f;
            Pi[m * PLD + c] = Si[r] * 0.125f;
        }
        __syncthreads();

        // exp(s - rowmax) in place: 2048 elements / 128 threads
#pragma unroll
        for (int j = 0; j < 16; ++j) {
            const int e     = t * 16 + j;
            const int plane = e >> 10;
            const int row   = (e & 1023) >> 5;
            const int col   = e & 31;
            float* Pp = plane ? Pi : Pr;
            const float mx = plane ? rmax_i[row] : rmax_r[row];
            Pp[row * PLD + col] = __expf(Pp[row * PLD + col] - mx);
        }
        __syncthreads();

        // kick off async V staging (overwrites K tile; all K reads done above),
        // and overlap the row-sum scan with the in-flight copies.
        stage_plane_async(Vr + base + (size_t)(kt * KB) * D_, KVr, t);
        stage_plane_async(Vi + base + (size_t)(kt * KB) * D_, KVi, t);
        if (t < 2 * QB) {
            const float* src = (t < QB) ? (Pr + t * PLD) : (Pi + (t - QB) * PLD);
            float s = 0.0f;
            for (int c = 0; c < KB; ++c) s += src[c];
            runsum += s;
        }
        wait_async0();
        __syncthreads();

        // P.V accumulation; V read straight from swizzled [key][64] layout
#pragma unroll
        for (int ks = 0; ks < KB / 4; ++ks) {
            const int kc = ks * 4 + half * 2;    // key index (pair kc, kc+1)
            v2f apr = lds2(Pr + (smt * 16 + lrow) * PLD + kc);
            v2f api = lds2(Pi + (smt * 16 + lrow) * PLD + kc);
#pragma unroll
            for (int ni = 0; ni < 2; ++ni) {
                const int dh  = snt * 32 + ni * 16 + lrow;
                const int chv = dh >> 1, o = dh & 1;
                const int a0 = kc * 64       + 2 * (chv ^ kc)       + o;
                const int a1 = (kc + 1) * 64 + 2 * (chv ^ (kc + 1)) + o;
                v2f bvr = (v2f){KVr[a0], KVr[a1]};
                v2f bvi = (v2f){KVi[a0], KVi[a1]};
                Arr[ni] = wmma_f32(apr, bvr, Arr[ni]);
                Aii[ni] = wmma_f32(api, bvi, Aii[ni]);
                Ari[ni] = wmma_f32(apr, bvi, Ari[ni]);
                Air[ni] = wmma_f32(api, bvr, Air[ni]);
            }
        }
        __syncthreads();
    }
    if (t < QB)            rsum_r[t]      = runsum;
    else if (t < 2 * QB)   rsum_i[t - QB] = runsum;
    __syncthreads();

    // ---- finalize: O_r = (Pr.Vr)/lr - (Pi.Vi)/li ; O_i = (Pr.Vi)/lr + (Pi.Vr)/li
#pragma unroll
    for (int ni = 0; ni < 2; ++ni) {
        const int dcol = snt * 32 + ni * 16 + lrow;
#pragma unroll
        for (int r = 0; r < 8; ++r) {
            const int m  = smt * 16 + r + half * 8;
            const float lr = rsum_r[m];
            const float li = rsum_i[m];
            const size_t addr = base + (size_t)(qb * QB + m) * D_ + dcol;
            Og_r[addr] = Arr[ni][r] / lr - Aii[ni][r] / li;
            Og_i[addr] = Ari[ni][r] / lr + Air[ni][r] / li;
        }
    }
}

// ---------------------------------------------------------------------------
// Kernel 3: residual (+ original complex q) and dual LayerNorm, interleave out.
// One block per (b,s) row; 256 threads x 2 elements.
// ---------------------------------------------------------------------------
__global__ __launch_bounds__(256)
void ln_kernel(const float* __restrict__ Og_r, const float* __restrict__ Og_i,
               const float* __restrict__ qin,
               const float* __restrict__ g_r, const float* __restrict__ b_r,
               const float* __restrict__ g_i, const float* __restrict__ b_i,
               float* __restrict__ out)
{
    __shared__ float red_a[256], red_b[256];
    const int    t  = threadIdx.x;
    const size_t rb = (size_t)blockIdx.x * D_;
    const int d0 = t * 2, d1 = t * 2 + 1;

    const float xr0 = Og_r[rb + d0] + qin[(rb + d0) * 2];
    const float xi0 = Og_i[rb + d0] + qin[(rb + d0) * 2 + 1];
    const float xr1 = Og_r[rb + d1] + qin[(rb + d1) * 2];
    const float xi1 = Og_i[rb + d1] + qin[(rb + d1) * 2 + 1];

    red_a[t] = xr0 + xr1;
    red_b[t] = xi0 + xi1;
    __syncthreads();
    for (int off = 128; off > 0; off >>= 1) {
        if (t < off) { red_a[t] += red_a[t + off]; red_b[t] += red_b[t + off]; }
        __syncthreads();
    }
    const float mur = red_a[0] * (1.0f / D_);
    const float mui = red_b[0] * (1.0f / D_);
    __syncthreads();

    const float dr0 = xr0 - mur, dr1 = xr1 - mur;
    const float di0 = xi0 - mui, di1 = xi1 - mui;
    red_a[t] = dr0 * dr0 + dr1 * dr1;
    red_b[t] = di0 * di0 + di1 * di1;
    __syncthreads();
    for (int off = 128; off > 0; off >>= 1) {
        if (t < off) { red_a[t] += red_a[t + off]; red_b[t] += red_b[t + off]; }
        __syncthreads();
    }
    const float rvr = rsqrtf(red_a[0] * (1.0f / D_) + EPS_);
    const float rvi = rsqrtf(red_b[0] * (1.0f / D_) + EPS_);

    out[(rb + d0) * 2 + 0] = dr0 * rvr * g_r[d0] + b_r[d0];
    out[(rb + d0) * 2 + 1] = di0 * rvi * g_i[d0] + b_i[d0];
    out[(rb + d1) * 2 + 0] = dr1 * rvr * g_r[d1] + b_r[d1];
    out[(rb + d1) * 2 + 1] = di1 * rvi * g_i[d1] + b_i[d1];
}

// ---------------------------------------------------------------------------
extern "C" void kernel_launch(void* const* d_in, const int* in_sizes, int n_in,
                              void* d_out, int out_size, void* d_ws, size_t ws_size,
                              hipStream_t stream)
{
    const float* q  = (const float*)d_in[0];
    const float* k  = (const float*)d_in[1];
    const float* v  = (const float*)d_in[2];
    const float* Wq = (const float*)d_in[3];
    const float* bq = (const float*)d_in[4];
    const float* Wk = (const float*)d_in[5];
    const float* bk = (const float*)d_in[6];
    const float* Wv = (const float*)d_in[7];
    const float* bv = (const float*)d_in[8];
    const float* gr = (const float*)d_in[9];
    const float* br = (const float*)d_in[10];
    const float* gi = (const float*)d_in[11];
    const float* bi = (const float*)d_in[12];

    float* ws = (float*)d_ws;
    const size_t plane = (size_t)B_ * S_ * D_;   // 2,097,152 floats
    float* Qr = ws + 0 * plane;  float* Qi = ws + 1 * plane;
    float* Kr = ws + 2 * plane;  float* Ki = ws + 3 * plane;
    float* Vr = ws + 4 * plane;  float* Vi = ws + 5 * plane;
    float* Or_ = ws + 6 * plane; float* Oi_ = ws + 7 * plane;

    dim3 pgrid(B_ * S_ / 128, D_ / 64);
    proj_kernel<<<pgrid, 256, 0, stream>>>(q, Wq, bq, Qr, Qi);
    proj_kernel<<<pgrid, 256, 0, stream>>>(k, Wk, bk, Kr, Ki);
    proj_kernel<<<pgrid, 256, 0, stream>>>(v, Wv, bv, Vr, Vi);

    attn_kernel<<<B_ * H_ * (S_ / QB), 128, 0, stream>>>(Qr, Qi, Kr, Ki, Vr, Vi,
                                                         Or_, Oi_);

    ln_kernel<<<B_ * S_, 256, 0, stream>>>(Or_, Oi_, q, gr, br, gi, bi,
                                           (float*)d_out);
}